// Multi_OICR_83923660963981
// MI455X (gfx1250) — compile-verified
//
#include <hip/hip_runtime.h>
#include <math.h>

typedef float v2f __attribute__((ext_vector_type(2)));
typedef float v8f __attribute__((ext_vector_type(8)));

#define BSZ   8
#define PN    2048
#define CIN   512
#define NROWS (BSZ * PN)   // 16384
#define WSTRIDE 516        // 512 + 4 pad floats: bank-conflict-free ds_load_b64

// -------- workspace layout (in floats) --------
// [0,        65536)  y        (16384 x 4)
// [65536,    81920)  w        (16384)
// [81920,    81924)  imbalance(4)
// at byte offset (81924*4): int cnt[24], int amax[24]
#define WS_Y    0
#define WS_W    65536
#define WS_IMB  81920
#define WS_INT  81924

// ---------------------------------------------------------------------------
// init: zero imbalance accumulators + loss output slot
// ---------------------------------------------------------------------------
__global__ void k_init(float* ws, float* out_loss) {
    int t = threadIdx.x;
    if (t < 4) ws[WS_IMB + t] = 0.0f;
    if (t == 0) *out_loss = 0.0f;
}

// ---------------------------------------------------------------------------
// per-(batch,class) stats: count of s>0.5 and first-argmax of s
// grid = (3 classes, 8 batches), 256 threads
// ---------------------------------------------------------------------------
__global__ void k_stats(const float* __restrict__ pre, float* ws) {
    int c = blockIdx.x;
    int b = blockIdx.y;
    int t = threadIdx.x;
    __shared__ float sval[256];
    __shared__ int   sidx[256];
    __shared__ int   scnt[256];

    float best = -1e30f; int bidx = 0; int cnt = 0;
    for (int j = t; j < PN; j += 256) {
        float s = pre[((long)b * PN + j) * 4 + c];
        if (s > 0.5f) cnt++;
        if (s > best) { best = s; bidx = j; }   // strict > keeps first max
    }
    sval[t] = best; sidx[t] = bidx; scnt[t] = cnt;
    __syncthreads();
    for (int off = 128; off > 0; off >>= 1) {
        if (t < off) {
            scnt[t] += scnt[t + off];
            float v2 = sval[t + off]; int i2 = sidx[t + off];
            if (v2 > sval[t] || (v2 == sval[t] && i2 < sidx[t])) {
                sval[t] = v2; sidx[t] = i2;
            }
        }
        __syncthreads();
    }
    if (t == 0) {
        int* ip = (int*)(ws + WS_INT);
        ip[b * 3 + c]      = scnt[0];
        ip[24 + b * 3 + c] = sidx[0];
    }
}

// ---------------------------------------------------------------------------
// GEMM (16384x512 @ 512x4, N padded to 16) via V_WMMA_F32_16X16X4_F32,
// then per-row softmax -> logit. One wave = one 16-row tile; 8 waves/block.
// fc_w staged transposed + zero-padded to 16 cols in LDS so every lane does
// an unconditional ds_load_b64 for its B fragment (no EXEC masking in loop).
// grid = 128 blocks x 256 threads (128 rows per block).
// ---------------------------------------------------------------------------
__global__ void k_gemm_softmax(const float* __restrict__ in,
                               const float* __restrict__ fcw,
                               const float* __restrict__ fcb,
                               float* __restrict__ logit) {
    __shared__ float lds_wT[16 * WSTRIDE];   // (n, k) layout, rows 4..15 zero
    __shared__ float lds_xr[128 * 4];

    int t = threadIdx.x;
    for (int idx = t; idx < 16 * WSTRIDE; idx += 256) lds_wT[idx] = 0.0f;
    __syncthreads();
    for (int idx = t; idx < CIN * 4; idx += 256) {
        int k = idx >> 2, n = idx & 3;            // fcw is (k, n) row-major
        lds_wT[n * WSTRIDE + k] = fcw[idx];
    }
    __syncthreads();

    int wave = t >> 5;
    int lane = t & 31;
    int hlf  = lane >> 4;       // 0: K=(k,k+1)   1: K=(k+2,k+3)
    int l    = lane & 15;       // A: row M=l     B/D: column N=l
    long row0 = (long)blockIdx.x * 128 + wave * 16;
    const float* arow = in + (row0 + l) * CIN + hlf * 2;
    const float* brow = lds_wT + l * WSTRIDE + hlf * 2;

    v8f acc = {};
    for (int k = 0; k < CIN; k += 4) {
        v2f a;
        a.x = arow[k];
        a.y = arow[k + 1];
        v2f bm;
        bm.x = brow[k];
        bm.y = brow[k + 1];
#if __has_builtin(__builtin_amdgcn_wmma_f32_16x16x4_f32)
        acc = __builtin_amdgcn_wmma_f32_16x16x4_f32(
            /*neg_a=*/false, a, /*neg_b=*/false, bm,
            /*c_mod=*/(short)0, acc, /*reuse_a=*/false, /*reuse_b=*/false);
#else
#warning "wmma_f32_16x16x4_f32 builtin missing: scalar fallback (no WMMA emitted)"
        for (int r = 0; r < 8; ++r) acc[r] += a.x * bm.x + a.y * bm.y;
#endif
    }

    // D layout: VGPR r -> M = hlf*8 + r, N = l (16x16 f32 C/D striping)
    if (l < 4) {
        float bias = fcb[l];
        #pragma unroll
        for (int r = 0; r < 8; ++r) {
            int rowL = wave * 16 + hlf * 8 + r;
            lds_xr[rowL * 4 + l] = acc[r] + bias;
        }
    }
    __syncthreads();

    if (t < 128) {
        float x0 = lds_xr[t * 4 + 0], x1 = lds_xr[t * 4 + 1];
        float x2 = lds_xr[t * 4 + 2], x3 = lds_xr[t * 4 + 3];
        float m  = fmaxf(fmaxf(x0, x1), fmaxf(x2, x3));
        float e0 = expf(x0 - m), e1 = expf(x1 - m);
        float e2 = expf(x2 - m), e3 = expf(x3 - m);
        float inv = 1.0f / (e0 + e1 + e2 + e3);
        long row = (long)blockIdx.x * 128 + t;
        logit[row * 4 + 0] = e0 * inv;
        logit[row * 4 + 1] = e1 * inv;
        logit[row * 4 + 2] = e2 * inv;
        logit[row * 4 + 3] = e3 * inv;
    }
}

// ---------------------------------------------------------------------------
// assignment: per (batch, i) masked IoU max + first-argmax for 3 classes.
// rois (32KB) + scores (32KB) for the batch staged into LDS with CDNA5
// async global->LDS b128 copies (ASYNCcnt), then scanned from LDS.
// smem layout: [0,32768) float4 rois ; [32768,65536) float4 scores.
// grid = (8 i-chunks, 8 batches) x 256 threads; thread = one proposal i.
// ---------------------------------------------------------------------------
__global__ void k_assign(const float* __restrict__ pre,
                         const float* __restrict__ labels,
                         const float* __restrict__ rois,
                         float* ws) {
    __shared__ __align__(16) char smem[65536];
    const float4* sroi = (const float4*)smem;
    const float*  ss4  = (const float*)(smem + 32768);

    int b = blockIdx.y;
    int t = threadIdx.x;

    // async stage: each lane copies 16B chunks straight to LDS (no VGPR data)
    for (int j = t; j < PN; j += 256) {
        unsigned roff = (unsigned)(j * 16);
        const float4* gro = (const float4*)rois + (long)b * PN + j;
        asm volatile("global_load_async_to_lds_b128 %0, %1, off"
                     :: "v"(roff), "v"(gro) : "memory");
        unsigned soff = (unsigned)(32768 + j * 16);
        const float4* gsc = (const float4*)pre + (long)b * PN + j;
        asm volatile("global_load_async_to_lds_b128 %0, %1, off"
                     :: "v"(soff), "v"(gsc) : "memory");
    }
    asm volatile("s_wait_asynccnt 0x0" ::: "memory");
    __syncthreads();

    const int* ip = (const int*)(ws + WS_INT);
    int i = blockIdx.x * 256 + t;
    float4 ri = sroi[i];
    float areai = (ri.z - ri.x) * (ri.w - ri.y);

    bool en0 = labels[b * 4 + 0] != 0.0f;
    bool en1 = labels[b * 4 + 1] != 0.0f;
    bool en2 = labels[b * 4 + 2] != 0.0f;
    bool th0 = ip[b * 3 + 0] > 1, th1 = ip[b * 3 + 1] > 1, th2 = ip[b * 3 + 2] > 1;
    int am0 = ip[24 + b * 3 + 0], am1 = ip[24 + b * 3 + 1], am2 = ip[24 + b * 3 + 2];

    float v0 = -2.0f, v1 = -2.0f, v2 = -2.0f;
    int   j0 = 0,     j1 = 0,     j2 = 0;

    for (int j = 0; j < PN; ++j) {
        float4 rj = sroi[j];
        float lx = fmaxf(ri.x, rj.x), ly = fmaxf(ri.y, rj.y);
        float rx = fminf(ri.z, rj.z), ry = fminf(ri.w, rj.w);
        float iw = fmaxf(rx - lx, 0.0f), ih = fmaxf(ry - ly, 0.0f);
        float inter = iw * ih;
        float areaj = (rj.z - rj.x) * (rj.w - rj.y);
        float iou = inter / (areai + areaj - inter);
        float s0 = ss4[j * 4 + 0], s1 = ss4[j * 4 + 1], s2 = ss4[j * 4 + 2];
        bool sel0 = en0 && (th0 ? (s0 > 0.5f) : (j == am0));
        bool sel1 = en1 && (th1 ? (s1 > 0.5f) : (j == am1));
        bool sel2 = en2 && (th2 ? (s2 > 0.5f) : (j == am2));
        float m0 = sel0 ? iou : -2.0f;
        float m1 = sel1 ? iou : -2.0f;
        float m2 = sel2 ? iou : -2.0f;
        if (m0 > v0) { v0 = m0; j0 = j; }      // strict > => first argmax
        if (m1 > v1) { v1 = m1; j1 = j; }
        if (m2 > v2) { v2 = m2; j2 = j; }
    }

    // running best across classes (in class order, strict >)
    float runI = -1.0f, runw = 1.0f;
    float c0 = ss4[j0 * 4 + 0], c1 = ss4[j1 * 4 + 1], c2 = ss4[j2 * 4 + 2];
    if (v0 > runI) { runw = c0; runI = v0; }
    if (v1 > runI) { runw = c1; runI = v1; }
    if (v2 > runI) { runw = c2; runI = v2; }

    float y0 = (v0 > 0.5f) ? 1.0f : 0.0f;
    float y1 = (v1 > 0.5f) ? 1.0f : 0.0f;
    float y2 = (v2 > 0.5f) ? 1.0f : 0.0f;
    float y3 = (y0 + y1 + y2 > 0.0f) ? 0.0f : 1.0f;

    long row = (long)b * PN + i;
    ws[WS_Y + row * 4 + 0] = y0;
    ws[WS_Y + row * 4 + 1] = y1;
    ws[WS_Y + row * 4 + 2] = y2;
    ws[WS_Y + row * 4 + 3] = y3;
    ws[WS_W + row] = runw;

    // imbalance = y.sum(0): warp-reduce then one atomic per lane0 (wave32)
    float r0 = y0, r1 = y1, r2 = y2, r3 = y3;
    for (int off = 16; off > 0; off >>= 1) {
        r0 += __shfl_down(r0, off, 32);
        r1 += __shfl_down(r1, off, 32);
        r2 += __shfl_down(r2, off, 32);
        r3 += __shfl_down(r3, off, 32);
    }
    if ((t & 31) == 0) {
        atomicAdd(&ws[WS_IMB + 0], r0);
        atomicAdd(&ws[WS_IMB + 1], r1);
        atomicAdd(&ws[WS_IMB + 2], r2);
        atomicAdd(&ws[WS_IMB + 3], r3);
    }
}

// ---------------------------------------------------------------------------
// focal loss reduction -> scalar. grid = 64 blocks x 256 threads (1 row each).
// ---------------------------------------------------------------------------
__global__ void k_loss(const float* __restrict__ logit,
                       const float* __restrict__ labels,
                       const float* __restrict__ ws,
                       float* out_loss) {
    __shared__ float red[256];
    int t = threadIdx.x;
    long row = (long)blockIdx.x * 256 + t;
    int b = (int)(row / PN);
    float wrow = ws[WS_W + row];
    float acc = 0.0f;
    #pragma unroll
    for (int c = 0; c < 4; ++c) {
        float lg = logit[row * 4 + c];
        float p  = fminf(fmaxf(lg, 1e-7f), 1.0f - 1e-7f);
        float y  = ws[WS_Y + row * 4 + c];
        float om = 1.0f - p;
        float fl = -y * logf(p) * om * om;                 // GAMMA = 2
        float lab = (c == 3) ? 1.0f : labels[b * 4 + c];
        float wd  = 10.0f * expf(lg) * (1.0f - lab) + lab;
        acc += wrow * fl / (ws[WS_IMB + c] + 1e-7f) * wd;
    }
    red[t] = acc;
    __syncthreads();
    for (int off = 128; off > 0; off >>= 1) {
        if (t < off) red[t] += red[t + off];
        __syncthreads();
    }
    if (t == 0) atomicAdd(out_loss, red[0] * (1.0f / (float)BSZ));
}

// ---------------------------------------------------------------------------
extern "C" void kernel_launch(void* const* d_in, const int* in_sizes, int n_in,
                              void* d_out, int out_size, void* d_ws, size_t ws_size,
                              hipStream_t stream) {
    const float* inputs = (const float*)d_in[0];   // (16384, 512)
    const float* fc_w   = (const float*)d_in[1];   // (512, 4)
    const float* fc_b   = (const float*)d_in[2];   // (4,)
    const float* pre    = (const float*)d_in[3];   // (8, 2048, 4)
    const float* labels = (const float*)d_in[4];   // (8, 4)
    const float* rois   = (const float*)d_in[5];   // (8, 2048, 4)
    float* out   = (float*)d_out;                  // logit (65536) + loss (1)
    float* ws    = (float*)d_ws;
    float* loss  = out + NROWS * 4;

    k_init<<<1, 64, 0, stream>>>(ws, loss);
    k_stats<<<dim3(3, BSZ), 256, 0, stream>>>(pre, ws);
    k_gemm_softmax<<<NROWS / 128, 256, 0, stream>>>(inputs, fc_w, fc_b, out);
    k_assign<<<dim3(PN / 256, BSZ), 256, 0, stream>>>(pre, labels, rois, ws);
    k_loss<<<NROWS / 256, 256, 0, stream>>>(out, labels, ws, loss);
}